// CRFlayer_72688026517606
// MI455X (gfx1250) — compile-verified
//
#include <hip/hip_runtime.h>

// CRF forward (logsumexp recursion) on CDNA5 via exact-f32 WMMA:
//   step:  alpha[b,j] <- emit[t,b,j] + m_b + log( sum_i exp(alpha[b,i]-m_b) * expT[i,j] )
// expT = exp(transitions) precomputed once; inner sum = (16xK)@(Kx16) f32 WMMA chain.
//
// Grid: 32 blocks x 128 threads (4 waves). Block owns 16 batch rows; wave w owns
// j-columns [16w,16w+16). B operand (expT tiles) is loop-invariant -> registers.
//
// Native v_exp_f32 / v_log_f32 used directly: the WMMA result d satisfies
// d >= exp(0)*min(expT) ~= 0.5 (row-max guarantees a unit term), so the libm
// denormal guard around log is provably dead weight on the 511-step chain.

#define T_ 512
#define B_ 512
#define N_ 64

#define LOG2E_F 1.4426950408889634f
#define LN2_F   0.6931471805599453f

typedef __attribute__((ext_vector_type(2))) float v2f;
typedef __attribute__((ext_vector_type(8))) float v8f;

__device__ __forceinline__ float fast_exp(float x) {
    return __builtin_amdgcn_exp2f(x * LOG2E_F);     // v_exp_f32 computes 2^x
}
__device__ __forceinline__ float fast_log(float x) {
    return __builtin_amdgcn_logf(x) * LN2_F;        // v_log_f32 computes log2(x)
}

__launch_bounds__(128)
__global__ void crf_forward_kernel(const float* __restrict__ emit,
                                   const float* __restrict__ trans,
                                   const float* __restrict__ strans,
                                   const float* __restrict__ etrans,
                                   const unsigned char* __restrict__ mask,
                                   float* __restrict__ partials)
{
    __shared__ float expT[N_ * N_];   // 16 KB: exp(transitions), row-major
    __shared__ float e_lds[16 * 68];  // exp(alpha - rowmax) staging; stride 68 -> conflict-free A reads
    __shared__ float red[4 * 16];     // cross-wave row reductions (max / sum)

    const int tid   = threadIdx.x;
    const int wid   = tid >> 5;       // wave id = j-block (0..3)
    const int lane  = tid & 31;
    const int m     = lane & 15;      // N-position within 16-wide tile / M-row for A operand
    const int half  = lane >> 4;      // 0/1: selects rows r / r+8 (C layout), K offset (A/B layout)
    const int jbase = wid * 16;
    const int j     = jbase + m;      // this lane's state column
    const int base  = blockIdx.x * 16; // batch-row base for this block

    // ---- precompute exp(transitions) ----
    for (int idx = tid; idx < N_ * N_; idx += blockDim.x)
        expT[idx] = fast_exp(trans[idx]);
    __syncthreads();

    // ---- hoist B operands (K x 16 tiles of expT), loop-invariant over time ----
    // B 4x16 layout: VGPR0 = rows K=0 (lanes 0-15) / K=2 (lanes 16-31); VGPR1 = K=1 / K=3.
    v2f Breg[16];
#pragma unroll
    for (int kb = 0; kb < 16; ++kb) {
        int k0 = kb * 4 + 2 * half;
        Breg[kb].x = expT[(k0 + 0) * N_ + j];
        Breg[kb].y = expT[(k0 + 1) * N_ + j];
    }

    // ---- alpha0 = emit[0] + strans, held in WMMA C layout:
    //      acc[r] = alpha[base + r + 8*half][j]
    v8f acc;
    {
        float st = strans[j];
#pragma unroll
        for (int r = 0; r < 8; ++r) {
            int row = r + 8 * half;
            acc[r] = emit[(size_t)(base + row) * N_ + j] + st;
        }
    }

    // ---- time recursion ----
    for (int t = 1; t < T_; ++t) {
        // row max over all 64 states: shuffle-reduce within the 16-lane half,
        // then combine the 4 waves' partials through LDS.
        float pm[8];
#pragma unroll
        for (int r = 0; r < 8; ++r) {
            float v = acc[r];
            v = fmaxf(v, __shfl_xor(v, 1, 32));
            v = fmaxf(v, __shfl_xor(v, 2, 32));
            v = fmaxf(v, __shfl_xor(v, 4, 32));
            v = fmaxf(v, __shfl_xor(v, 8, 32));
            pm[r] = v;                      // uniform within each 16-lane half
        }
        if (m == 0) {
#pragma unroll
            for (int r = 0; r < 8; ++r) red[wid * 16 + r + 8 * half] = pm[r];
        }
        __syncthreads();
        float rm[8];
#pragma unroll
        for (int r = 0; r < 8; ++r) {
            int row = r + 8 * half;
            float v = red[0 * 16 + row];
            v = fmaxf(v, red[1 * 16 + row]);
            v = fmaxf(v, red[2 * 16 + row]);
            v = fmaxf(v, red[3 * 16 + row]);
            rm[r] = v;
        }
        // e = exp(alpha - rowmax) -> LDS (row-major, padded)
#pragma unroll
        for (int r = 0; r < 8; ++r) {
            int row = r + 8 * half;
            e_lds[row * 68 + j] = fast_exp(acc[r] - rm[r]);
        }
        __syncthreads();

        // d = e @ expT for this wave's 16-column block: 16 chained f32 WMMAs (K=64).
        // A 16x4 layout: lane m holds K = {2*half, 2*half+1} in VGPRs {0,1}.
        v8f d = {};
#pragma unroll
        for (int kb = 0; kb < 16; ++kb) {
            int k0 = kb * 4 + 2 * half;
            v2f a;
            a.x = e_lds[m * 68 + k0 + 0];
            a.y = e_lds[m * 68 + k0 + 1];
            d = __builtin_amdgcn_wmma_f32_16x16x4_f32(false, a, false, Breg[kb],
                                                      (short)0, d, false, false);
        }

        // alpha <- mask ? emit + rowmax + log(d) : alpha
#pragma unroll
        for (int r = 0; r < 8; ++r) {
            int row = r + 8 * half;
            int b   = base + row;
            unsigned char mk = mask[t * B_ + b];
            float em = emit[((size_t)t * B_ + b) * N_ + j];
            float sc = em + rm[r] + fast_log(d[r]);
            acc[r] = mk ? sc : acc[r];
        }
        // No trailing barrier needed: next iteration's red[] write is ordered
        // after this iteration's red[] reads by the e_lds barrier above, and
        // e_lds reuse is ordered by next iteration's first barrier.
    }

    // ---- per-block logZ partial: sum over 16 rows of logsumexp_j(alpha + etrans) ----
    float et = etrans[j];
    float val[8];
#pragma unroll
    for (int r = 0; r < 8; ++r) val[r] = acc[r] + et;

    float pm2[8];
#pragma unroll
    for (int r = 0; r < 8; ++r) {
        float v = val[r];
        v = fmaxf(v, __shfl_xor(v, 1, 32));
        v = fmaxf(v, __shfl_xor(v, 2, 32));
        v = fmaxf(v, __shfl_xor(v, 4, 32));
        v = fmaxf(v, __shfl_xor(v, 8, 32));
        pm2[r] = v;
    }
    if (m == 0) {
#pragma unroll
        for (int r = 0; r < 8; ++r) red[wid * 16 + r + 8 * half] = pm2[r];
    }
    __syncthreads();
    float rm2[8];
#pragma unroll
    for (int r = 0; r < 8; ++r) {
        int row = r + 8 * half;
        float v = red[0 * 16 + row];
        v = fmaxf(v, red[1 * 16 + row]);
        v = fmaxf(v, red[2 * 16 + row]);
        v = fmaxf(v, red[3 * 16 + row]);
        rm2[r] = v;
    }
    __syncthreads();   // before reusing red[] for sums
    float ps[8];
#pragma unroll
    for (int r = 0; r < 8; ++r) {
        float s = fast_exp(val[r] - rm2[r]);
        s += __shfl_xor(s, 1, 32);
        s += __shfl_xor(s, 2, 32);
        s += __shfl_xor(s, 4, 32);
        s += __shfl_xor(s, 8, 32);
        ps[r] = s;
    }
    if (m == 0) {
#pragma unroll
        for (int r = 0; r < 8; ++r) red[wid * 16 + r + 8 * half] = ps[r];
    }
    __syncthreads();

    if (wid == 0 && m == 0) {          // lanes 0 and 16: rows 0-7 / 8-15
        float psum = 0.f;
#pragma unroll
        for (int r = 0; r < 8; ++r) {
            int row = r + 8 * half;
            float tot = red[0 * 16 + row] + red[1 * 16 + row] +
                        red[2 * 16 + row] + red[3 * 16 + row];
            psum += rm2[r] + fast_log(tot);
        }
        psum += __shfl_xor(psum, 16, 32);
        if (tid == 0) partials[blockIdx.x] = psum;
    }
}

// ---- gold path score: one thread per batch element (deterministic, tiny) ----
__global__ void crf_gold_kernel(const float* __restrict__ emit,
                                const float* __restrict__ trans,
                                const float* __restrict__ strans,
                                const float* __restrict__ etrans,
                                const int* __restrict__ target,
                                const unsigned char* __restrict__ mask,
                                float* __restrict__ gold_out)
{
    int b = blockIdx.x * blockDim.x + threadIdx.x;
    if (b >= B_) return;
    float s = 0.f;
    int prev = target[b];
    int cnt = 0;
    if (mask[b]) { s += emit[(size_t)b * N_ + prev]; cnt = 1; }
    for (int t = 1; t < T_; ++t) {
        int tg = target[t * B_ + b];
        if (mask[t * B_ + b]) {
            s += emit[((size_t)t * B_ + b) * N_ + tg] + trans[prev * N_ + tg];
            cnt++;
        }
        prev = tg;
    }
    s += strans[target[b]];
    int ends = cnt - 1;                // lengths >= 1 so cnt >= 1
    s += etrans[target[ends * B_ + b]];
    gold_out[b] = s;
}

// ---- final fixed-order reduction: out = sum(logZ partials) - sum(gold) ----
__global__ void crf_final_kernel(const float* __restrict__ ws, float* __restrict__ out)
{
    if (blockIdx.x == 0 && threadIdx.x == 0) {
        float lz = 0.f;
        for (int i = 0; i < 32; ++i) lz += ws[i];
        float g = 0.f;
        for (int b = 0; b < B_; ++b) g += ws[32 + b];
        out[0] = lz - g;
    }
}

extern "C" void kernel_launch(void* const* d_in, const int* in_sizes, int n_in,
                              void* d_out, int out_size, void* d_ws, size_t ws_size,
                              hipStream_t stream)
{
    const float*         emit   = (const float*)d_in[0];          // (T,B,N) f32
    const float*         trans  = (const float*)d_in[1];          // (N,N)   f32
    const float*         strans = (const float*)d_in[2];          // (N,)    f32
    const float*         etrans = (const float*)d_in[3];          // (N,)    f32
    const int*           target = (const int*)d_in[4];            // (T,B)   i32
    const unsigned char* mask   = (const unsigned char*)d_in[5];  // (T,B)   bool (1 byte)
    float* ws = (float*)d_ws;   // [0,32): per-block logZ partials; [32,32+B): per-b gold

    crf_forward_kernel<<<B_ / 16, 128, 0, stream>>>(emit, trans, strans, etrans, mask, ws);
    crf_gold_kernel<<<(B_ + 255) / 256, 256, 0, stream>>>(emit, trans, strans, etrans,
                                                          target, mask, ws + 32);
    crf_final_kernel<<<1, 32, 0, stream>>>(ws, (float*)d_out);
}